// Attention_29746943492190
// MI455X (gfx1250) — compile-verified
//
#include <hip/hip_runtime.h>
#include <hip/hip_bf16.h>

// CDNA5 / gfx1250. Whole net collapses to:
//   G_b = X_b X_b^T (128x128)                      [kernel 1, fp32 WMMA]
//   scores = scale * Wq G Wk^T ; softmax           [kernel 2, small]
//   A_b = stack_h(attn_h Wv_h)/n ; M_b = Wout A_b  [kernels 2+3, small]
//   y_b = M_b X_b + b_out                          [kernel 4, fp32 WMMA]
// Memory-bound: ~84 MB HBM traffic -> fp32 WMMA (16x16x4) keeps exact precision.

typedef __attribute__((ext_vector_type(2))) float v2f;
typedef __attribute__((ext_vector_type(8))) float v8f;

#define NTOT 32768   // 32*32*32
#define CDIM 128
#define NHEADS 8
#define DHEAD 64

__device__ __forceinline__ v8f wmma4(v2f a, v2f b, v8f c) {
    // V_WMMA_F32_16X16X4_F32 : D = A(16x4) * B(4x16) + C
    return __builtin_amdgcn_wmma_f32_16x16x4_f32(
        false, a, false, b, (short)0, c, false, false);
}

// ---------------------------------------------------------------------------
// Kernel 1: Gram matrix G_b = X_b X_b^T via fp32 WMMA + atomic accumulation.
// grid = B * (NTOT/256) blocks of 256 threads (8 waves).
// Each block handles a 256-column slab; each wave owns one 16x128 row strip.
// ---------------------------------------------------------------------------
#define LDA 36   // padded LDS stride (words): conflict-free column reads, 16B rows

__global__ void gram_kernel(const float* __restrict__ x, float* __restrict__ G) {
    __shared__ float tile[CDIM * LDA];              // 128 x 32 staged columns
    const int blk   = blockIdx.x;
    const int b     = blk >> 7;                     // 128 chunks per batch
    const int chunk = blk & 127;
    const float* Xb = x + (size_t)b * CDIM * NTOT + (size_t)chunk * 256;

    const int t    = threadIdx.x;
    const int lane = t & 31;
    const int w    = t >> 5;
    const int r16  = lane & 15;
    const int koff = (lane >> 4) << 1;              // 0 or 2
    const int arow = (w * 16 + r16) * LDA;          // A-frag LDS row base (runtime w)

    v8f acc[8];
    v8f zero = {};
#pragma unroll
    for (int i = 0; i < 8; ++i) acc[i] = zero;

    for (int s = 0; s < 8; ++s) {                   // 8 stages of 32 columns
        // stage 128 rows x 32 cols into LDS (float4, coalesced per row-segment)
        const int row = t >> 1;
        const int c0  = (t & 1) * 16;
        const float* src = Xb + (size_t)row * NTOT + s * 32 + c0;
        float* dst = &tile[row * LDA + c0];
#pragma unroll
        for (int q = 0; q < 4; ++q)
            *(float4*)(dst + q * 4) = *(const float4*)(src + q * 4);
        __syncthreads();

#pragma unroll
        for (int k = 0; k < 32; k += 4) {
            // A-fragment for this wave's row strip: loaded directly from LDS
            // (runtime address -- NO dynamic register-array indexing / cndmask tree)
            v2f fa;
            {
                const float* p = &tile[arow + k + koff];
                fa.x = p[0];
                fa.y = p[1];
            }
            v2f f[8];                               // B-fragments, col-tiles 0..7
#pragma unroll
            for (int tt = 0; tt < 8; ++tt) {
                const float* p = &tile[(tt * 16 + r16) * LDA + k + koff];
                f[tt].x = p[0];
                f[tt].y = p[1];
            }
#pragma unroll
            for (int ct = 0; ct < 8; ++ct)
                acc[ct] = wmma4(fa, f[ct], acc[ct]);
        }
        __syncthreads();
    }

    // accumulate partial Gram into global scratch
    float* Gb = G + b * CDIM * CDIM;
    const int hi = (lane >> 4) << 3;
#pragma unroll
    for (int ct = 0; ct < 8; ++ct) {
        const int col = ct * 16 + r16;
#pragma unroll
        for (int v = 0; v < 8; ++v) {
            const int row = w * 16 + v + hi;
            atomicAdd(&Gb[row * CDIM + col], acc[ct][v]);
        }
    }
}

// ---------------------------------------------------------------------------
// Kernel 2: per (b,h): T1 = Wq_h * G (G symmetric -> row-row dots),
// scores = scale * T1 * Wk_h^T, softmax, A_T[b][c][h*64+i] = (attn Wv_h)/n.
// grid = B*HEADS = 16 blocks of 256 threads.
// ---------------------------------------------------------------------------
__global__ void head_kernel(const float* __restrict__ G,
                            const float* __restrict__ w_qkv,
                            float* __restrict__ A_T) {
    __shared__ float T1[64 * 132];                  // 33.8 KB
    __shared__ float sc[64 * 66];                   // 16.9 KB (scores -> attn)

    const int b = blockIdx.x >> 3;
    const int h = blockIdx.x & 7;
    const int t = threadIdx.x;
    const float* Gb = G + b * CDIM * CDIM;

    const int i  = t >> 2;                          // 0..63 output row
    const int cb = (t & 3) * 32;                    // 32-wide column block

    // Phase 1: T1[i, c2] = dot(Wq[h*64+i,:], G[c2,:])   (symmetry of G)
    {
        float acc[32];
#pragma unroll
        for (int cc = 0; cc < 32; ++cc) acc[cc] = 0.f;
        const float* wq = w_qkv + (size_t)(h * DHEAD + i) * CDIM;
        for (int c1 = 0; c1 < CDIM; c1 += 4) {
            const float4 q4 = *(const float4*)(wq + c1);
#pragma unroll
            for (int cc = 0; cc < 32; ++cc) {
                const float4 g4 = *(const float4*)(Gb + (cb + cc) * CDIM + c1);
                acc[cc] += q4.x * g4.x + q4.y * g4.y + q4.z * g4.z + q4.w * g4.w;
            }
        }
#pragma unroll
        for (int cc = 0; cc < 32; ++cc) T1[i * 132 + cb + cc] = acc[cc];
    }
    __syncthreads();

    // Phase 2: scores[i,j] = scale * dot(T1[i,:], Wk[h*64+j,:])
    {
        const int jb = (t & 3) * 16;
        float sa[16];
#pragma unroll
        for (int jj = 0; jj < 16; ++jj) sa[jj] = 0.f;
        const float* wkb = w_qkv + (size_t)(512 + h * DHEAD + jb) * CDIM;
        for (int c2 = 0; c2 < CDIM; c2 += 4) {
            const float4 t4 = *(const float4*)(&T1[i * 132 + c2]);
#pragma unroll
            for (int jj = 0; jj < 16; ++jj) {
                const float4 k4 = *(const float4*)(wkb + jj * CDIM + c2);
                sa[jj] += t4.x * k4.x + t4.y * k4.y + t4.z * k4.z + t4.w * k4.w;
            }
        }
        const float scale = 0.125f;                 // 64^-0.5
#pragma unroll
        for (int jj = 0; jj < 16; ++jj) sc[i * 66 + jb + jj] = sa[jj] * scale;
    }
    __syncthreads();

    // Phase 3: row softmax (64 rows)
    if (t < 64) {
        float m = -1e30f;
        for (int j = 0; j < 64; ++j) m = fmaxf(m, sc[t * 66 + j]);
        float s = 0.f;
        for (int j = 0; j < 64; ++j) {
            const float e = __expf(sc[t * 66 + j] - m);
            sc[t * 66 + j] = e;
            s += e;
        }
        const float inv = 1.f / s;
        for (int j = 0; j < 64; ++j) sc[t * 66 + j] *= inv;
    }
    __syncthreads();

    // Phase 4: A_T[b][c][h*64+i] = (1/n) * sum_j attn[i,j] * Wv[h*64+j, c]
    {
        float4 av[8];
        const float4 z4 = {0.f, 0.f, 0.f, 0.f};
#pragma unroll
        for (int q = 0; q < 8; ++q) av[q] = z4;
        const float* wv = w_qkv + (size_t)(1024 + h * DHEAD) * CDIM + cb;
        for (int j = 0; j < 64; ++j) {
            const float a = sc[i * 66 + j];
            const float* wr = wv + j * CDIM;
#pragma unroll
            for (int q = 0; q < 8; ++q) {
                const float4 v4 = *(const float4*)(wr + q * 4);
                av[q].x += a * v4.x; av[q].y += a * v4.y;
                av[q].z += a * v4.z; av[q].w += a * v4.w;
            }
        }
        const float invn = 1.0f / (float)NTOT;
        float* at = A_T + (size_t)b * CDIM * 512;
#pragma unroll
        for (int q = 0; q < 8; ++q) {
            const int c = cb + q * 4;
            const int hc = h * DHEAD + i;
            at[(c + 0) * 512 + hc] = av[q].x * invn;
            at[(c + 1) * 512 + hc] = av[q].y * invn;
            at[(c + 2) * 512 + hc] = av[q].z * invn;
            at[(c + 3) * 512 + hc] = av[q].w * invn;
        }
    }
}

// ---------------------------------------------------------------------------
// Kernel 3: M[b][o][c] = dot(w_out[o,:], A_T[b][c][:])   (512-long dots)
// grid = 64 blocks of 256 threads, one element each.
// ---------------------------------------------------------------------------
__global__ void proj_kernel(const float* __restrict__ w_out,
                            const float* __restrict__ A_T,
                            float* __restrict__ M) {
    const int e = blockIdx.x * 256 + threadIdx.x;   // 0..16383
    const int b = e >> 13;
    const int r = e & 8191;
    const int o = r >> 7;
    const int c = r & 127;
    const float* wo = w_out + (size_t)o * 512;
    const float* at = A_T + (size_t)b * CDIM * 512 + (size_t)c * 512;
    float acc = 0.f;
    for (int k = 0; k < 512; k += 4) {
        const float4 a4 = *(const float4*)(wo + k);
        const float4 b4 = *(const float4*)(at + k);
        acc += a4.x * b4.x + a4.y * b4.y + a4.z * b4.z + a4.w * b4.w;
    }
    M[(size_t)b * 8192 + o * CDIM + c] = acc;
}

// ---------------------------------------------------------------------------
// Kernel 4: y_b = M_b (64x128) @ X_b (128xN) + b_out, via fp32 WMMA.
// grid = B * (NTOT/256) blocks of 256 threads; wave -> 64 rows x 32 cols.
// ---------------------------------------------------------------------------
#define LDM 132  // padded stride: conflict-free column reads, 16B-aligned rows

__global__ void out_kernel(const float* __restrict__ x,
                           const float* __restrict__ M,
                           const float* __restrict__ b_out,
                           float* __restrict__ y) {
    __shared__ float m_s[64 * LDM];                 // 33.8 KB
    const int b     = blockIdx.x >> 7;
    const int chunk = blockIdx.x & 127;
    const int t     = threadIdx.x;

    // stage M_b into LDS
    const float* Mb = M + (size_t)b * 8192;
#pragma unroll
    for (int q = 0; q < 8; ++q) {
        const int fid  = t + q * 256;               // float4 id, 0..2047
        const int row  = fid >> 5;
        const int colf = fid & 31;
        *(float4*)&m_s[row * LDM + colf * 4] =
            *(const float4*)(Mb + row * CDIM + colf * 4);
    }
    __syncthreads();

    const int lane = t & 31;
    const int w    = t >> 5;
    const int r16  = lane & 15;
    const int koff = (lane >> 4) << 1;
    const int hi   = (lane >> 4) << 3;
    const size_t n0 = (size_t)chunk * 256 + w * 32;
    const float* Xb = x + (size_t)b * CDIM * NTOT;

    v8f acc[8];                                     // [mt*2 + ct]
    v8f zero = {};
#pragma unroll
    for (int i = 0; i < 8; ++i) acc[i] = zero;

    for (int k = 0; k < CDIM; k += 4) {
        v2f bf[2];
#pragma unroll
        for (int ct = 0; ct < 2; ++ct) {            // B[kk,j] = X[k+kk, n]
            const size_t base = (size_t)(k + koff) * NTOT + n0 + ct * 16 + r16;
            bf[ct].x = Xb[base];
            bf[ct].y = Xb[base + NTOT];
        }
        v2f af[4];
#pragma unroll
        for (int mt = 0; mt < 4; ++mt) {            // A[m,kk] = M[mt*16+m, k+kk]
            const float* p = &m_s[(mt * 16 + r16) * LDM + k + koff];
            af[mt].x = p[0];
            af[mt].y = p[1];
        }
#pragma unroll
        for (int mt = 0; mt < 4; ++mt)
#pragma unroll
            for (int ct = 0; ct < 2; ++ct)
                acc[mt * 2 + ct] = wmma4(af[mt], bf[ct], acc[mt * 2 + ct]);
    }

    float* Yb = y + (size_t)b * 64 * NTOT;
#pragma unroll
    for (int mt = 0; mt < 4; ++mt) {
#pragma unroll
        for (int v = 0; v < 8; ++v) {
            const int o  = mt * 16 + v + hi;
            const float bo = b_out[o];
#pragma unroll
            for (int ct = 0; ct < 2; ++ct) {
                const size_t n = n0 + ct * 16 + r16;
                Yb[(size_t)o * NTOT + n] = acc[mt * 2 + ct][v] + bo;
            }
        }
    }
}

// ---------------------------------------------------------------------------
extern "C" void kernel_launch(void* const* d_in, const int* in_sizes, int n_in,
                              void* d_out, int out_size, void* d_ws, size_t ws_size,
                              hipStream_t stream) {
    const float* x     = (const float*)d_in[0];   // (2,128,32,32,32)
    const float* w_qkv = (const float*)d_in[1];   // (1536,128)
    const float* w_out = (const float*)d_in[2];   // (64,512)
    const float* b_out = (const float*)d_in[3];   // (64,)
    float* y = (float*)d_out;                     // (2,64,32,32,32)

    float* G   = (float*)d_ws;                    // B*128*128   = 32768 floats
    float* A_T = G + 2 * CDIM * CDIM;             // B*128*512   = 131072 floats
    float* M   = A_T + 2 * CDIM * 512;            // B*64*128    = 16384 floats

    // zero Gram accumulators (atomic targets)
    hipMemsetAsync(G, 0, (size_t)2 * CDIM * CDIM * sizeof(float), stream);

    gram_kernel<<<dim3(2 * 128), dim3(256), 0, stream>>>(x, G);
    head_kernel<<<dim3(16),      dim3(256), 0, stream>>>(G, w_qkv, A_T);
    proj_kernel<<<dim3(64),      dim3(256), 0, stream>>>(w_out, A_T, M);
    out_kernel<<<dim3(2 * 128),  dim3(256), 0, stream>>>(x, M, b_out, y);
}